// LightGCN_27891517620688
// MI455X (gfx1250) — compile-verified
//
#include <hip/hip_runtime.h>

#define N_USERS  100000
#define N_ITEMS  50000
#define N_NODES  (N_USERS + N_ITEMS)
#define DIM      64
#define NNZ      6400000
#define BATCH    4096
#define N_LAYERS 3

typedef __attribute__((ext_vector_type(2))) float v2f;
typedef __attribute__((ext_vector_type(8))) float v8f;

// Guaranteed single-instruction f32 atomic add at device scope (RMW executes
// at L2; non-returning form uses STOREcnt, drained by S_ENDPGM's implicit
// wait-idle).
__device__ __forceinline__ void atomic_fadd_dev(float* p, float v) {
    asm volatile("global_atomic_add_f32 %0, %1, off scope:SCOPE_DEV"
                 :: "v"(p), "v"(v)
                 : "memory");
}

// ---------------------------------------------------------------------------
// SpMM scatter: one wave (32 lanes) per edge; lane l owns dims [2l, 2l+1].
// Gather of cur[col] is a fully coalesced 256B row read (L2-resident table),
// scatter is 2 x global_atomic_add_f32 per lane.
// ---------------------------------------------------------------------------
__global__ void spmm_scatter(const int* __restrict__ erow,
                             const int* __restrict__ ecol,
                             const float* __restrict__ eval,
                             const float* __restrict__ cur,
                             float* __restrict__ nxt) {
    const int lane   = threadIdx.x & 31;
    const int wave   = (int)((blockIdx.x * blockDim.x + threadIdx.x) >> 5);
    const int nwaves = (int)((gridDim.x * blockDim.x) >> 5);
    for (int e = wave; e < NNZ; e += nwaves) {
        const int   r = erow[e];
        const int   c = ecol[e];
        const float v = eval[e];
        const float2 x = *(const float2*)(cur + (size_t)c * DIM + lane * 2);
        float* dst = nxt + (size_t)r * DIM + lane * 2;
        atomic_fadd_dev(dst + 0, v * x.x);
        atomic_fadd_dev(dst + 1, v * x.y);
    }
}

// acc = cur = concat(user_emb, item_emb), as float4 streams.
__global__ void init_concat(const float4* __restrict__ ue,
                            const float4* __restrict__ ie,
                            float4* __restrict__ acc,
                            float4* __restrict__ cur) {
    const int i = blockIdx.x * blockDim.x + threadIdx.x;
    const int total = N_NODES * (DIM / 4);
    if (i >= total) return;
    const int usplit = N_USERS * (DIM / 4);
    float4 v = (i < usplit) ? ue[i] : ie[i - usplit];
    acc[i] = v;
    cur[i] = v;
}

__global__ void zero_buf(float4* __restrict__ p) {
    const int i = blockIdx.x * blockDim.x + threadIdx.x;
    if (i < N_NODES * (DIM / 4)) p[i] = make_float4(0.f, 0.f, 0.f, 0.f);
}

__global__ void acc_add(float4* __restrict__ acc, const float4* __restrict__ nxt) {
    const int i = blockIdx.x * blockDim.x + threadIdx.x;
    if (i >= N_NODES * (DIM / 4)) return;
    float4 a = acc[i], b = nxt[i];
    a.x += b.x; a.y += b.y; a.z += b.z; a.w += b.w;
    acc[i] = a;
}

// ---------------------------------------------------------------------------
// Batched dot via V_WMMA_F32_16X16X4_F32:  gamma = (U .* I) x ones
// One wave handles a tile of 16 batch rows. A-operand layout (16x4 f32):
//   lanes 0-15 : M = lane,    VGPR0=K0, VGPR1=K1
//   lanes16-31 : M = lane-16, VGPR0=K2, VGPR1=K3
// 16 WMMA steps of K=4 cover DIM=64. D column N=0 lives in lane 0 (M=0..7)
// and lane 16 (M=8..15) across the 8 accumulator VGPRs.
// ---------------------------------------------------------------------------
__global__ void dot_wmma(const float* __restrict__ acc,
                         const int* __restrict__ users,
                         const int* __restrict__ items,
                         float* __restrict__ out) {
    const int lane = threadIdx.x & 31;
    const int wave = (int)((blockIdx.x * blockDim.x + threadIdx.x) >> 5);
    const int row0 = wave * 16;            // 16 batch rows per wave
    const int m    = lane & 15;
    const int b    = row0 + m;

    const int u  = users[b];
    const int it = items[b];
    const float* __restrict__ pu = acc + (size_t)u * DIM;
    const float* __restrict__ pi = acc + (size_t)(N_USERS + it) * DIM;

    const int kbase = (lane >> 4) * 2;     // 0 for lanes 0-15, 2 for 16-31

    v8f c = {};
    v2f ones; ones.x = 1.0f; ones.y = 1.0f;

    #pragma unroll
    for (int step = 0; step < 16; ++step) {
        const int k = step * 4 + kbase;
        v2f a;
        a.x = pu[k]     * pi[k];
        a.y = pu[k + 1] * pi[k + 1];
        c = __builtin_amdgcn_wmma_f32_16x16x4_f32(
                /*neg_a=*/false, a, /*neg_b=*/false, ones,
                /*c_mod=*/(short)0, c, /*reuse_a=*/false, /*reuse_b=*/false);
    }

    // light_out = acc/4 for both sides  ->  gamma = dot(accU, accI) / 16
    const float s = 1.0f / 16.0f;
    if (lane == 0) {
        #pragma unroll
        for (int r = 0; r < 8; ++r) out[row0 + r] = c[r] * s;
    } else if (lane == 16) {
        #pragma unroll
        for (int r = 0; r < 8; ++r) out[row0 + 8 + r] = c[r] * s;
    }
}

extern "C" void kernel_launch(void* const* d_in, const int* in_sizes, int n_in,
                              void* d_out, int out_size, void* d_ws, size_t ws_size,
                              hipStream_t stream) {
    const float* ue    = (const float*)d_in[0];
    const float* ie    = (const float*)d_in[1];
    const int*   erow  = (const int*)d_in[2];
    const int*   ecol  = (const int*)d_in[3];
    const float* eval  = (const float*)d_in[4];
    const int*   users = (const int*)d_in[5];
    const int*   items = (const int*)d_in[6];
    float*       out   = (float*)d_out;

    const size_t embBytes = (size_t)N_NODES * DIM * sizeof(float); // 38.4 MB
    char* base = (char*)d_ws;
    float* acc = (float*)(base);
    float* b0  = (float*)(base + embBytes);
    float* b1  = (float*)(base + 2 * embBytes);

    const int ELEM4  = N_NODES * (DIM / 4);
    const int eblk   = (ELEM4 + 255) / 256;

    init_concat<<<eblk, 256, 0, stream>>>((const float4*)ue, (const float4*)ie,
                                          (float4*)acc, (float4*)b0);

    float* cur = b0;
    float* nxt = b1;
    for (int l = 0; l < N_LAYERS; ++l) {
        zero_buf<<<eblk, 256, 0, stream>>>((float4*)nxt);
        spmm_scatter<<<8192, 256, 0, stream>>>(erow, ecol, eval, cur, nxt);
        acc_add<<<eblk, 256, 0, stream>>>((float4*)acc, (const float4*)nxt);
        float* t = cur; cur = nxt; nxt = t;
    }

    // BATCH/16 = 256 waves, 8 waves per 256-thread block -> 32 blocks
    dot_wmma<<<BATCH / 16 / 8, 256, 0, stream>>>(acc, users, items, out);
}